// SemanticEncoderVQVAE_28106265985619
// MI455X (gfx1250) — compile-verified
//
#include <hip/hip_runtime.h>
#include <math.h>

// ---------------------------------------------------------------------------
// Fused VQ-VAE semantic encoder for gfx1250 (MI455X), wave32 + f32 WMMA.
//   h   = elu(c @ W1 + b1)            [B,200]
//   z_e = h @ W2 + b2                 [B,20]
//   d_j = ||z||^2 + ||cb_j||^2 - 2 z.cb_j ; idx = argmin_j ; z_q = cb[idx]
//   mu  = z_e @ Wmu + bmu ; lv = z_e @ Wlv + blv
//   qloss = mean((q - z)^2) = mean(min-dist) ; closs = 0.25 * qloss
// All matrix math via V_WMMA_F32_16X16X4_F32. Weights live in LDS, swizzled
// [ktile][n][4] so each B-fragment is one conflict-free ds_load_b64.
// ---------------------------------------------------------------------------

typedef float v2f __attribute__((ext_vector_type(2)));
typedef float v8f __attribute__((ext_vector_type(8)));

#define WMMA4(a, b, c) \
  __builtin_amdgcn_wmma_f32_16x16x4_f32(false, (a), false, (b), (short)0, (c), false, false)

// ---- LDS layout (float offsets), dynamic shared memory ----
constexpr int OFF_W1  = 0;                      // 25 ktiles * 208 cols * 4 = 20800
constexpr int OFF_W2  = OFF_W1  + 20800;        // 50 * 32 * 4 = 6400
constexpr int OFF_WMU = OFF_W2  + 6400;         // 5 * 32 * 4 = 640
constexpr int OFF_WLV = OFF_WMU + 640;          // 640
constexpr int OFF_CBT = OFF_WLV + 640;          // 5 * 512 * 4 = 10240
constexpr int OFF_B1  = OFF_CBT + 10240;        // 208
constexpr int OFF_B2  = OFF_B1  + 208;          // 32
constexpr int OFF_BMU = OFF_B2  + 32;           // 32
constexpr int OFF_BLV = OFF_BMU + 32;           // 32
constexpr int OFF_CBN = OFF_BLV + 32;           // 512 codebook norms
constexpr int HSTR    = 210;                    // h tile row stride (bank-skewed, even)
constexpr int OFF_H   = OFF_CBN + 512;          // 8 waves * 16*210 = 26880
constexpr int OFF_Z   = OFF_H   + 8 * 16 * HSTR;// 8 waves * 16*20  = 2560
constexpr int OFF_MND = OFF_Z   + 8 * 320;      // 8 waves * 16
constexpr int OFF_MNJ = OFF_MND + 8 * 16;       // 8 waves * 16 (int)
constexpr int LDS_FLOATS = OFF_MNJ + 8 * 16;    // ~276.9 KB < 320 KB/WGP

__global__ __launch_bounds__(256) void vqvae_fused_kernel(
    const float* __restrict__ c,  const float* __restrict__ W1, const float* __restrict__ b1,
    const float* __restrict__ W2, const float* __restrict__ b2, const float* __restrict__ cb,
    const float* __restrict__ Wmu, const float* __restrict__ bmu,
    const float* __restrict__ Wlv, const float* __restrict__ blv,
    float* __restrict__ out_mu, float* __restrict__ out_lv,
    float* __restrict__ out_zq, float* __restrict__ out_ze,
    float* __restrict__ ws, int B) {
  extern __shared__ float smem[];
  const int tid  = threadIdx.x;
  const int wave = tid >> 5;
  const int lane = tid & 31;
  const int lo   = lane & 15;      // column / M-row selector within half
  const int hi   = lane >> 4;      // half selector
  const int klo  = hi * 2;         // K offset of this half inside a K=4 step

  // ---------------- stage weights into LDS (swizzled [ktile][n][4]) --------
  for (int f = tid; f < 20800; f += 256) {            // W1: 100x200 -> pad N 208
    int kk = f & 3, rem = f >> 2, n = rem % 208, kt = rem / 208;
    smem[OFF_W1 + f] = (n < 200) ? W1[(kt * 4 + kk) * 200 + n] : 0.f;
  }
  for (int f = tid; f < 6400; f += 256) {             // W2: 200x20 -> pad N 32
    int kk = f & 3, rem = f >> 2, n = rem & 31, kt = rem >> 5;
    smem[OFF_W2 + f] = (n < 20) ? W2[(kt * 4 + kk) * 20 + n] : 0.f;
  }
  for (int f = tid; f < 640; f += 256) {              // Wmu / Wlv: 20x20 -> 32
    int kk = f & 3, rem = f >> 2, n = rem & 31, kt = rem >> 5;
    smem[OFF_WMU + f] = (n < 20) ? Wmu[(kt * 4 + kk) * 20 + n] : 0.f;
    smem[OFF_WLV + f] = (n < 20) ? Wlv[(kt * 4 + kk) * 20 + n] : 0.f;
  }
  for (int f = tid; f < 10240; f += 256) {            // codebook^T: K=20, N=512
    int kk = f & 3, rem = f >> 2, j = rem & 511, kt = rem >> 9;
    smem[OFF_CBT + f] = cb[j * 20 + kt * 4 + kk];
  }
  for (int n = tid; n < 208; n += 256) smem[OFF_B1 + n] = (n < 200) ? b1[n] : 0.f;
  if (tid < 32) {
    smem[OFF_B2  + tid] = (tid < 20) ? b2[tid]  : 0.f;
    smem[OFF_BMU + tid] = (tid < 20) ? bmu[tid] : 0.f;
    smem[OFF_BLV + tid] = (tid < 20) ? blv[tid] : 0.f;
  }
  for (int j = tid; j < 512; j += 256) {              // codebook row norms
    float s = 0.f;
    for (int k = 0; k < 20; ++k) { float t = cb[j * 20 + k]; s += t * t; }
    smem[OFF_CBN + j] = s;
  }
  __syncthreads();

  float* hS = &smem[OFF_H + wave * 16 * HSTR];
  float* zS = &smem[OFF_Z + wave * 320];
  float* mnD = &smem[OFF_MND + wave * 16];
  int*   mnJ = (int*)&smem[OFF_MNJ + wave * 16];

  const int numTiles = B >> 4;
  for (int tile = blockIdx.x * 8 + wave; tile < numTiles; tile += gridDim.x * 8) {
    const int rowBase = tile * 16;

    // ---- load A fragments of c tile: 16 rows x 100 K, f32 16x16x4 layout ----
    v2f a1[25];
    const float* crow = c + (size_t)(rowBase + lo) * 100 + klo;
    #pragma unroll
    for (int kt = 0; kt < 25; ++kt) a1[kt] = *(const v2f*)(crow + kt * 4);

    // ---- GEMM1 + bias + ELU, write h tile (row-major) to LDS ----
    for (int nt = 0; nt < 13; ++nt) {
      v8f acc = {};
      #pragma unroll
      for (int kt = 0; kt < 25; ++kt) {
        v2f b = *(const v2f*)&smem[OFF_W1 + ((kt * 208 + nt * 16 + lo) << 2) + klo];
        acc = WMMA4(a1[kt], b, acc);
      }
      float bias = smem[OFF_B1 + nt * 16 + lo];
      #pragma unroll
      for (int v = 0; v < 8; ++v) {
        float x = acc[v] + bias;
        x = (x > 0.f) ? x : (__expf(x) - 1.f);          // ELU(alpha=1)
        hS[(v + 8 * hi) * HSTR + nt * 16 + lo] = x;
      }
    }
    asm volatile("s_wait_dscnt 0" ::: "memory");

    // ---- GEMM2: z_e = h @ W2 + b2 (N padded 20->32), write z tile + out ----
    for (int nt = 0; nt < 2; ++nt) {
      v8f acc = {};
      #pragma unroll
      for (int kt = 0; kt < 50; ++kt) {
        v2f a = *(const v2f*)&hS[lo * HSTR + kt * 4 + klo];
        v2f b = *(const v2f*)&smem[OFF_W2 + ((kt * 32 + nt * 16 + lo) << 2) + klo];
        acc = WMMA4(a, b, acc);
      }
      int col = nt * 16 + lo;
      float bias = smem[OFF_B2 + col];
      #pragma unroll
      for (int v = 0; v < 8; ++v) {
        if (col < 20) {
          float zv = acc[v] + bias;
          int row = v + 8 * hi;
          zS[row * 20 + col] = zv;
          out_ze[(size_t)(rowBase + row) * 20 + col] = zv;
        }
      }
    }
    asm volatile("s_wait_dscnt 0" ::: "memory");

    // ---- A fragments of z_e (K = 20 exactly = 5 WMMA K-steps) ----
    v2f az[5];
    #pragma unroll
    for (int kt = 0; kt < 5; ++kt) az[kt] = *(const v2f*)&zS[lo * 20 + kt * 4 + klo];

    // ---- mu / logvar heads ----
    for (int nt = 0; nt < 2; ++nt) {
      v8f am = {}, al = {};
      #pragma unroll
      for (int kt = 0; kt < 5; ++kt) {
        v2f bm = *(const v2f*)&smem[OFF_WMU + ((kt * 32 + nt * 16 + lo) << 2) + klo];
        v2f bl = *(const v2f*)&smem[OFF_WLV + ((kt * 32 + nt * 16 + lo) << 2) + klo];
        am = WMMA4(az[kt], bm, am);
        al = WMMA4(az[kt], bl, al);
      }
      int col = nt * 16 + lo;
      if (col < 20) {
        float bmv = smem[OFF_BMU + col], blv_ = smem[OFF_BLV + col];
        #pragma unroll
        for (int v = 0; v < 8; ++v) {
          int row = v + 8 * hi;
          out_mu[(size_t)(rowBase + row) * 20 + col] = am[v] + bmv;
          out_lv[(size_t)(rowBase + row) * 20 + col] = al[v] + blv_;
        }
      }
    }

    // ---- per-row ||z||^2, broadcast to C-fragment rows ----
    float zn = 0.f;
    #pragma unroll
    for (int k = 0; k < 20; ++k) { float t = zS[lo * 20 + k]; zn += t * t; }
    float znv[8];
    #pragma unroll
    for (int v = 0; v < 8; ++v) znv[v] = __shfl(zn, v + 8 * hi, 32);

    // ---- VQ: 32 codebook tiles of 16, running argmin per (row, lane-col) ----
    float bestD[8]; int bestJ[8];
    #pragma unroll
    for (int v = 0; v < 8; ++v) { bestD[v] = 3.4e38f; bestJ[v] = 0; }
    for (int jt = 0; jt < 32; ++jt) {
      v8f acc = {};
      #pragma unroll
      for (int kt = 0; kt < 5; ++kt) {
        v2f b = *(const v2f*)&smem[OFF_CBT + ((kt * 512 + jt * 16 + lo) << 2) + klo];
        acc = WMMA4(az[kt], b, acc);
      }
      float cn = smem[OFF_CBN + jt * 16 + lo];
      int j = jt * 16 + lo;
      #pragma unroll
      for (int v = 0; v < 8; ++v) {
        float d = znv[v] + cn - 2.f * acc[v];           // == sum((q - z)^2)
        if (d < bestD[v]) { bestD[v] = d; bestJ[v] = j; }  // first-min tiebreak
      }
    }
    // cross-lane argmin within each 16-lane half (rows 0-7 / 8-15)
    #pragma unroll
    for (int s = 8; s >= 1; s >>= 1) {
      #pragma unroll
      for (int v = 0; v < 8; ++v) {
        float od = __shfl_xor(bestD[v], s, 32);
        int   oj = __shfl_xor(bestJ[v], s, 32);
        if (od < bestD[v] || (od == bestD[v] && oj < bestJ[v])) {
          bestD[v] = od; bestJ[v] = oj;
        }
      }
    }
    if (lo == 0) {
      #pragma unroll
      for (int v = 0; v < 8; ++v) { mnD[v + 8 * hi] = bestD[v]; mnJ[v + 8 * hi] = bestJ[v]; }
    }
    asm volatile("s_wait_dscnt 0" ::: "memory");

    if (lane == 0) {            // tile partial for losses (min-dist == SSE)
      float s = 0.f;
      for (int m = 0; m < 16; ++m) s += mnD[m];
      ws[tile] = s;
    }
    // z_q = codebook[idx] : 16 rows x 20 cols = 320 elems, 10 per lane
    #pragma unroll
    for (int t = 0; t < 10; ++t) {
      int e = lane * 10 + t;
      int row = e / 20, col = e - row * 20;
      int j = mnJ[row];
      float q = smem[OFF_CBT + (((col >> 2) * 512 + j) << 2) + (col & 3)];
      out_zq[(size_t)(rowBase + row) * 20 + col] = q;
    }
  }
}

// Deterministic loss reduction: fixed per-thread strided order, then a fixed
// sequential sum of 256 partials by thread 0.
__global__ __launch_bounds__(256) void vq_loss_reduce(const float* __restrict__ ws,
                                                      float* __restrict__ out,
                                                      int n, float invCount) {
  __shared__ float p[256];
  float s = 0.f;
  for (int i = threadIdx.x; i < n; i += 256) s += ws[i];
  p[threadIdx.x] = s;
  __syncthreads();
  if (threadIdx.x == 0) {
    float t = 0.f;
    for (int i = 0; i < 256; ++i) t += p[i];
    float q = t * invCount;
    out[0] = q;           // quantization_loss = mean((q-z)^2)
    out[1] = 0.25f * q;   // commitment_loss
  }
}

extern "C" void kernel_launch(void* const* d_in, const int* in_sizes, int n_in,
                              void* d_out, int out_size, void* d_ws, size_t ws_size,
                              hipStream_t stream) {
  const float* c   = (const float*)d_in[0];
  const float* W1  = (const float*)d_in[1];
  const float* b1  = (const float*)d_in[2];
  const float* W2  = (const float*)d_in[3];
  const float* b2  = (const float*)d_in[4];
  const float* cb  = (const float*)d_in[5];
  const float* Wmu = (const float*)d_in[6];
  const float* bmu = (const float*)d_in[7];
  const float* Wlv = (const float*)d_in[8];
  const float* blv = (const float*)d_in[9];

  const int B = in_sizes[0] / 100;        // 262144
  const int numTiles = B / 16;            // 16384 16-row tiles
  float* out = (float*)d_out;
  float* ws  = (float*)d_ws;              // numTiles floats of partial SSE

  const size_t ldsBytes = (size_t)LDS_FLOATS * sizeof(float);  // ~277 KB
  (void)hipFuncSetAttribute((const void*)vqvae_fused_kernel,
                            hipFuncAttributeMaxDynamicSharedMemorySize, (int)ldsBytes);

  const int blocks = (numTiles + 7) / 8;  // 8 waves/WG, one 16-row tile each
  vqvae_fused_kernel<<<blocks, 256, ldsBytes, stream>>>(
      c, W1, b1, W2, b2, cb, Wmu, bmu, Wlv, blv,
      out,                               // mu
      out + (size_t)B * 20,              // logvar
      out + 2 * (size_t)B * 20,          // z_q
      out + 3 * (size_t)B * 20,          // z_e
      ws, B);
  vq_loss_reduce<<<1, 256, 0, stream>>>(ws, out + 4 * (size_t)B * 20,
                                        numTiles, 1.0f / ((float)B * 20.0f));
}